// CIGA_8847632630350
// MI455X (gfx1250) — compile-verified
//
#include <hip/hip_runtime.h>

// ---------------------------------------------------------------------------
// CIGA edge-scoring pipeline for MI455X (gfx1250, wave32, WMMA)
//   - 3-layer GCN (sym-normalized, self loops) : v_wmma_f32_16x16x4_f32 GEMMs
//     with weights staged in LDS (conflict-free padded layout)
//   - edge MLP  relu([E,128]@[128,256]) @ [256,1] : fused WMMA kernel,
//     Wa1 staged once per workgroup in 132KB dynamic LDS (8 waves share)
//   - per-graph top-k (ratio 0.25) : exact 4-pass radix select on key bits
// ---------------------------------------------------------------------------

typedef __attribute__((ext_vector_type(2))) float v2f;
typedef __attribute__((ext_vector_type(8))) float v8f;

#define GRAPHS 128
#define W_STRIDE 72     // 64 + 8 : (2*72) % 64 = 16 -> halves hit disjoint banks
#define WA1_STRIDE 264  // 256 + 8: (2*264) % 64 = 16 -> halves hit disjoint banks
#define WA1_SMEM_BYTES (128 * WA1_STRIDE * 4)

__device__ __forceinline__ v8f wmma_f32(v2f a, v2f b, v8f c) {
  // (neg_a, A, neg_b, B, c_mod, C, reuse_a, reuse_b)
  return __builtin_amdgcn_wmma_f32_16x16x4_f32(false, a, false, b, (short)0, c,
                                               false, false);
}

// order-preserving float -> uint key (larger float -> larger uint)
__device__ __forceinline__ unsigned keybits(float s) {
  unsigned u = __float_as_uint(s);
  return (u & 0x80000000u) ? ~u : (u | 0x80000000u);
}

// ---------------------------------------------------------------- utilities
__global__ void k_zero_i32(int* __restrict__ p, int n) {
  int i = blockIdx.x * blockDim.x + threadIdx.x;
  if (i < n) p[i] = 0;
}

__global__ void k_deg(const int* __restrict__ col, int* __restrict__ cnt, int E) {
  int i = blockIdx.x * blockDim.x + threadIdx.x;
  if (i < E) atomicAdd(&cnt[col[i]], 1);
}

__global__ void k_invdeg(const int* __restrict__ cnt, float* __restrict__ inv_sqrt,
                         float* __restrict__ inv_deg, int N) {
  int i = blockIdx.x * blockDim.x + threadIdx.x;
  if (i < N) {
    float d = 1.0f + (float)cnt[i];
    inv_sqrt[i] = rsqrtf(d);
    inv_deg[i] = 1.0f / d;
  }
}

__global__ void k_norme(const int* __restrict__ row, const int* __restrict__ col,
                        const float* __restrict__ inv_sqrt,
                        float* __restrict__ norme, int E) {
  int i = blockIdx.x * blockDim.x + threadIdx.x;
  if (i < E) norme[i] = inv_sqrt[row[i]] * inv_sqrt[col[i]];
}

// ------------------------------------------------- node GEMM: m = h@W + b
// One wave computes a 16x64 tile: 4 accumulators of 16x16, K=64 as 16 steps.
// W (64x64) staged in LDS with padded stride for conflict-free B reads.
__global__ __launch_bounds__(256) void k_gemm_node(
    const float* __restrict__ h, const float* __restrict__ W,
    const float* __restrict__ bias, float* __restrict__ m, int nrows) {
  __shared__ float sW[64 * W_STRIDE];

  // cooperative stage of W into LDS (coalesced)
  for (int idx = threadIdx.x; idx < 64 * 64; idx += 256) {
    int k = idx >> 6;
    int j = idx & 63;
    sW[k * W_STRIDE + j] = W[idx];
  }
  __syncthreads();

  int gtid = blockIdx.x * blockDim.x + threadIdx.x;
  int wave = gtid >> 5;
  int lane = threadIdx.x & 31;
  int r0 = wave << 4;
  if (r0 >= nrows) return;
  int mr = lane & 15;            // row (A/M) or col (B/N) handled by lane
  int kh = (lane >> 4) << 1;     // k sub-offset: 0 or 2

  v8f acc[4] = {};
  const float* arow = h + (size_t)(r0 + mr) * 64;
#pragma unroll
  for (int ks = 0; ks < 16; ++ks) {
    int kb = ks * 4 + kh;
    v2f a;
    a.x = arow[kb];
    a.y = arow[kb + 1];
#pragma unroll
    for (int c = 0; c < 4; ++c) {
      v2f b;
      b.x = sW[kb * W_STRIDE + c * 16 + mr];
      b.y = sW[(kb + 1) * W_STRIDE + c * 16 + mr];
      acc[c] = wmma_f32(a, b, acc[c]);
    }
  }
  int mhi = (lane >> 4) << 3;    // D layout: M = r + 8*(lane/16)
#pragma unroll
  for (int c = 0; c < 4; ++c) {
#pragma unroll
    for (int r = 0; r < 8; ++r) {
      int M = r + mhi;
      m[(size_t)(r0 + M) * 64 + c * 16 + mr] = acc[c][r] + bias[c * 16 + mr];
    }
  }
}

// h_next = m * inv_deg[node]   (self-loop term)
__global__ void k_selfloop(const float* __restrict__ m,
                           const float* __restrict__ inv_deg,
                           float* __restrict__ hnext, int total) {
  int i = blockIdx.x * blockDim.x + threadIdx.x;
  if (i < total) hnext[i] = m[i] * inv_deg[i >> 6];
}

// h_next[col] += m[row] * norm_e   (64 threads per edge; f32 atomics)
__global__ __launch_bounds__(256) void k_scatter(
    const float* __restrict__ m, const int* __restrict__ row,
    const int* __restrict__ col, const float* __restrict__ norme,
    float* __restrict__ hnext, int E) {
  int gid = blockIdx.x * blockDim.x + threadIdx.x;
  int e = gid >> 6;
  int f = gid & 63;
  if (e < E) {
    float v = m[(size_t)row[e] * 64 + f] * norme[e];
    atomicAdd(&hnext[(size_t)col[e] * 64 + f], v);
  }
}

__global__ void k_relu(float* __restrict__ h, int n) {
  int i = blockIdx.x * blockDim.x + threadIdx.x;
  if (i < n) h[i] = fmaxf(h[i], 0.0f);
}

// group[e] = batch[row[e]]; per-graph edge counts
__global__ void k_group(const int* __restrict__ row, const int* __restrict__ batch,
                        int* __restrict__ group, int* __restrict__ counts, int E) {
  int i = blockIdx.x * blockDim.x + threadIdx.x;
  if (i < E) {
    int g = batch[row[i]];
    group[i] = g;
    atomicAdd(&counts[g], 1);
  }
}

// ---------------------------------- fused edge MLP: score = relu(ER@Wa1+ba1)@Wa2+ba2
// One wave per 16 edges. A = concat(h[row],h[col]) 16x128 kept in 64 VGPRs.
// Wa1 (128x256) staged once per workgroup in dynamic LDS (132KB, padded rows);
// 16 N-tiles x 32 K-steps of v_wmma_f32_16x16x4_f32, ReLU*Wa2 epilogue fused.
__global__ __launch_bounds__(256) void k_edge_mlp(
    const float* __restrict__ h, const int* __restrict__ row,
    const int* __restrict__ col, const float* __restrict__ Wa1,
    const float* __restrict__ ba1, const float* __restrict__ Wa2,
    const float* __restrict__ ba2, float* __restrict__ score, int nedges) {
  extern __shared__ float sWa1[];  // 128 * WA1_STRIDE floats

  // cooperative stage of Wa1 into LDS (coalesced float4 reads)
  {
    const float4* src = (const float4*)Wa1;
    for (int idx = threadIdx.x; idx < (128 * 256) / 4; idx += 256) {
      float4 v = src[idx];
      int lin = idx * 4;
      int k = lin >> 8;        // /256
      int j = lin & 255;
      float* dst = &sWa1[k * WA1_STRIDE + j];
      dst[0] = v.x; dst[1] = v.y; dst[2] = v.z; dst[3] = v.w;
    }
  }
  __syncthreads();

  int gtid = blockIdx.x * blockDim.x + threadIdx.x;
  int wave = gtid >> 5;
  int lane = threadIdx.x & 31;
  int e0 = wave << 4;
  if (e0 >= nedges) return;
  int mr = lane & 15;
  int kh = (lane >> 4) << 1;
  int e = e0 + mr;

  const float* rrow = h + (size_t)row[e] * 64;
  const float* crow = h + (size_t)col[e] * 64;

  v2f afr[32];
#pragma unroll
  for (int ks = 0; ks < 16; ++ks) {
    int kb = ks * 4 + kh;
    afr[ks].x = rrow[kb];
    afr[ks].y = rrow[kb + 1];
    afr[ks + 16].x = crow[kb];
    afr[ks + 16].y = crow[kb + 1];
  }

  float saccum[8];
#pragma unroll
  for (int r = 0; r < 8; ++r) saccum[r] = 0.0f;

#pragma unroll 1
  for (int c = 0; c < 16; ++c) {
    v8f acc = {};
    int n = c * 16 + mr;  // hidden unit handled by this lane (N index)
#pragma unroll
    for (int ks = 0; ks < 32; ++ks) {
      int kb = ks * 4 + kh;  // K index into the 128-wide concat dimension
      v2f b;
      b.x = sWa1[kb * WA1_STRIDE + n];
      b.y = sWa1[(kb + 1) * WA1_STRIDE + n];
      acc = wmma_f32(afr[ks], b, acc);
    }
    float bb = ba1[n];
    float w2 = Wa2[n];
#pragma unroll
    for (int r = 0; r < 8; ++r) {
      float vv = acc[r] + bb;
      vv = vv > 0.0f ? vv : 0.0f;
      saccum[r] += vv * w2;
    }
  }

  float b2 = ba2[0];
#pragma unroll
  for (int r = 0; r < 8; ++r) {
    float v = saccum[r];
    v += __shfl_xor(v, 1, 16);
    v += __shfl_xor(v, 2, 16);
    v += __shfl_xor(v, 4, 16);
    v += __shfl_xor(v, 8, 16);
    if (mr == 0) {
      int M = r + ((lane >> 4) << 3);
      score[e0 + M] = v + b2;
    }
  }
}

// ------------------------------------------------ per-graph top-k (radix)
__global__ void k_initsel(const int* __restrict__ counts, int* __restrict__ kq,
                          unsigned* __restrict__ prefix, int* __restrict__ rem) {
  int g = threadIdx.x;
  if (g < GRAPHS) {
    int k = (counts[g] + 3) >> 2;  // ceil(0.25 * counts)
    kq[g] = k;
    prefix[g] = 0u;
    rem[g] = k;
  }
}

__global__ void k_hist(const float* __restrict__ score, const int* __restrict__ group,
                       const unsigned* __restrict__ prefix, int* __restrict__ hist,
                       int E, int b) {
  int i = blockIdx.x * blockDim.x + threadIdx.x;
  if (i >= E) return;
  unsigned u = keybits(score[i]);
  unsigned fm = (b == 3) ? 0u : (0xFFFFFFFFu << ((b + 1) * 8));
  int g = group[i];
  if ((u & fm) == prefix[g])
    atomicAdd(&hist[g * 256 + ((u >> (b * 8)) & 255)], 1);
}

__global__ void k_select(int* __restrict__ hist, unsigned* __restrict__ prefix,
                         int* __restrict__ rem, int b) {
  int g = threadIdx.x;
  if (g < GRAPHS) {
    int r = rem[g];
    int* hg = hist + g * 256;
    if (r > 0) {
      int cum = 0;
      for (int i = 255; i >= 0; --i) {
        int c = hg[i];
        cum += c;
        if (cum >= r) {
          rem[g] = r - (cum - c);
          prefix[g] |= ((unsigned)i) << (b * 8);
          break;
        }
      }
    }
    for (int i = 0; i < 256; ++i) hg[i] = 0;  // ready for next pass
  }
}

__global__ void k_final(const float* __restrict__ score, const int* __restrict__ group,
                        const int* __restrict__ kq, const unsigned* __restrict__ prefix,
                        const int* __restrict__ tiequota, int* __restrict__ tiecnt,
                        float* __restrict__ out, int E) {
  int i = blockIdx.x * blockDim.x + threadIdx.x;
  if (i >= E) return;
  float s = score[i];
  int g = group[i];
  int msk = 0;
  if (kq[g] > 0) {
    unsigned u = keybits(s);
    unsigned T = prefix[g];
    if (u > T) {
      msk = 1;
    } else if (u == T) {
      msk = (atomicAdd(&tiecnt[g], 1) < tiequota[g]) ? 1 : 0;
    }
  }
  out[(size_t)E + i] = msk ? s : 0.0f;
  out[(size_t)2 * E + i] = msk ? 0.0f : -s;
  out[(size_t)3 * E + i] = (float)msk;
}

// ---------------------------------------------------------------------------
extern "C" void kernel_launch(void* const* d_in, const int* in_sizes, int n_in,
                              void* d_out, int out_size, void* d_ws, size_t ws_size,
                              hipStream_t stream) {
  const float* x = (const float*)d_in[0];
  const int* edge_index = (const int*)d_in[1];
  const int* batch = (const int*)d_in[2];
  const float* W1 = (const float*)d_in[3];
  const float* b1 = (const float*)d_in[4];
  const float* W2 = (const float*)d_in[5];
  const float* b2 = (const float*)d_in[6];
  const float* W3 = (const float*)d_in[7];
  const float* b3 = (const float*)d_in[8];
  const float* Wa1 = (const float*)d_in[9];
  const float* ba1 = (const float*)d_in[10];
  const float* Wa2 = (const float*)d_in[11];
  const float* ba2 = (const float*)d_in[12];

  const int N = in_sizes[0] / 64;   // 50000
  const int E = in_sizes[1] / 2;    // 800000
  const int* row = edge_index;
  const int* col = edge_index + E;

  // ---- workspace carve-up (256B aligned) ----
  char* ws = (char*)d_ws;
  size_t off = 0;
  auto alloc = [&](size_t bytes) {
    void* p = ws + off;
    off = (off + bytes + 255) & ~(size_t)255;
    return p;
  };
  int* cnt = (int*)alloc((size_t)N * 4);
  float* inv_sqrt = (float*)alloc((size_t)N * 4);
  float* inv_deg = (float*)alloc((size_t)N * 4);
  float* norme = (float*)alloc((size_t)E * 4);
  int* group = (int*)alloc((size_t)E * 4);
  float* mbuf = (float*)alloc((size_t)N * 64 * 4);
  float* hA = (float*)alloc((size_t)N * 64 * 4);
  float* hB = (float*)alloc((size_t)N * 64 * 4);
  int* counts = (int*)alloc((size_t)GRAPHS * 4);
  int* kq = (int*)alloc((size_t)GRAPHS * 4);
  unsigned* prefix = (unsigned*)alloc((size_t)GRAPHS * 4);
  int* rem = (int*)alloc((size_t)GRAPHS * 4);
  int* tiecnt = (int*)alloc((size_t)GRAPHS * 4);
  int* hist = (int*)alloc((size_t)GRAPHS * 256 * 4);
  (void)ws_size;

  float* out = (float*)d_out;  // [0,E)=score, [E,2E)=causal, [2E,3E)=spu, [3E,4E)=mask

  const int TPB = 256;
  auto grid1 = [&](int n) { return dim3((unsigned)((n + TPB - 1) / TPB)); };

  // ---- init / zero ----
  k_zero_i32<<<grid1(N), TPB, 0, stream>>>(cnt, N);
  k_zero_i32<<<grid1(GRAPHS), TPB, 0, stream>>>(counts, GRAPHS);
  k_zero_i32<<<grid1(GRAPHS), TPB, 0, stream>>>(tiecnt, GRAPHS);
  k_zero_i32<<<grid1(GRAPHS * 256), TPB, 0, stream>>>(hist, GRAPHS * 256);

  // ---- degrees & normalization ----
  k_deg<<<grid1(E), TPB, 0, stream>>>(col, cnt, E);
  k_invdeg<<<grid1(N), TPB, 0, stream>>>(cnt, inv_sqrt, inv_deg, N);
  k_norme<<<grid1(E), TPB, 0, stream>>>(row, col, inv_sqrt, norme, E);

  // ---- 3 GCN layers ----
  const int nwaves = (N + 15) / 16;
  const dim3 ggrid((unsigned)((nwaves * 32 + TPB - 1) / TPB));
  const float* hin = x;
  const float* Ws[3] = {W1, W2, W3};
  const float* bs[3] = {b1, b2, b3};
  float* hbuf[2] = {hA, hB};
  for (int L = 0; L < 3; ++L) {
    float* hout = hbuf[L & 1];
    k_gemm_node<<<ggrid, TPB, 0, stream>>>(hin, Ws[L], bs[L], mbuf, N);
    k_selfloop<<<grid1(N * 64), TPB, 0, stream>>>(mbuf, inv_deg, hout, N * 64);
    k_scatter<<<grid1(E * 64), TPB, 0, stream>>>(mbuf, row, col, norme, hout, E);
    k_relu<<<grid1(N * 64), TPB, 0, stream>>>(hout, N * 64);
    hin = hout;
  }

  // ---- edge groups + counts ----
  k_group<<<grid1(E), TPB, 0, stream>>>(row, batch, group, counts, E);

  // ---- fused edge MLP -> score (row 0 of out) ----
  const int ewaves = (E + 15) / 16;
  const dim3 egrid((unsigned)((ewaves * 32 + TPB - 1) / TPB));
  k_edge_mlp<<<egrid, TPB, WA1_SMEM_BYTES, stream>>>(hin, row, col, Wa1, ba1, Wa2,
                                                     ba2, out, E);

  // ---- per-graph top-k radix select (4 passes, byte 3 -> 0) ----
  k_initsel<<<1, GRAPHS, 0, stream>>>(counts, kq, prefix, rem);
  for (int b = 3; b >= 0; --b) {
    k_hist<<<grid1(E), TPB, 0, stream>>>(out, group, prefix, hist, E, b);
    k_select<<<1, GRAPHS, 0, stream>>>(hist, prefix, rem, b);
  }

  // ---- final mask + causal/spurious weights ----
  k_final<<<grid1(E), TPB, 0, stream>>>(out, group, kq, prefix, rem, tiecnt, out, E);
}